// HeteroGNNLayer_54176717472255
// MI455X (gfx1250) — compile-verified
//
#include <hip/hip_runtime.h>
#include <hip/hip_bf16.h>

typedef __attribute__((ext_vector_type(2))) float v2f;
typedef __attribute__((ext_vector_type(8))) float v8f;

#define D 128
#define LDS_STRIDE 132   // pad to avoid 16-way LDS bank conflicts

// ---------------------------------------------------------------------------
// Zero-fill (float4 granularity; n4 = number of float4s)
// ---------------------------------------------------------------------------
__global__ __launch_bounds__(256) void zero4_kernel(float4* __restrict__ p, long n4) {
    long i = (long)blockIdx.x * blockDim.x + threadIdx.x;
    if (i < n4) p[i] = make_float4(0.f, 0.f, 0.f, 0.f);
}

// ---------------------------------------------------------------------------
// Edge scatter: one wave32 per edge; each lane moves a float4 of the row.
// aggI[i] += user_features[u];  aggU[u] += item_features[i];  degrees count.
// Uses native global_atomic_add_f32 (unsafeAtomicAdd).
// ---------------------------------------------------------------------------
__global__ __launch_bounds__(256)
void gnn_scatter_kernel(const float* __restrict__ UF, const float* __restrict__ IF,
                        const int* __restrict__ uidx, const int* __restrict__ iidx,
                        float* __restrict__ aggU, float* __restrict__ aggI,
                        float* __restrict__ degU, float* __restrict__ degI, int E)
{
    int t = blockIdx.x * 256 + threadIdx.x;
    int e = t >> 5;
    if (e >= E) return;
    int lane = threadIdx.x & 31;
    int u  = uidx[e];
    int it = iidx[e];

    const float4 fu = ((const float4*)(UF + (size_t)u  * D))[lane];
    const float4 fi = ((const float4*)(IF + (size_t)it * D))[lane];

    float* ai = aggI + (size_t)it * D + lane * 4;
    float* au = aggU + (size_t)u  * D + lane * 4;

    unsafeAtomicAdd(ai + 0, fu.x);
    unsafeAtomicAdd(ai + 1, fu.y);
    unsafeAtomicAdd(ai + 2, fu.z);
    unsafeAtomicAdd(ai + 3, fu.w);

    unsafeAtomicAdd(au + 0, fi.x);
    unsafeAtomicAdd(au + 1, fi.y);
    unsafeAtomicAdd(au + 2, fi.z);
    unsafeAtomicAdd(au + 3, fi.w);

    if (lane == 0) {
        unsafeAtomicAdd(degU + u, 1.0f);
        unsafeAtomicAdd(degI + it, 1.0f);
    }
}

// ---------------------------------------------------------------------------
// Fused dual-GEMM + bias + degree-scaled message bias, via V_WMMA_F32_16X16X4_F32.
//   out[row] = X[row] @ Wself.T + agg[row] @ Wmsg.T + bself + deg[row]*bmsg
// agg lives IN-PLACE in `out` (staged to LDS before overwrite).
// Block = 256 threads = 8 waves; block computes 16 rows x 128 cols;
// wave w computes the 16x16 tile at columns [16w, 16w+16).
// Requires rows % 16 == 0 (true: 100000, 50000).
// ---------------------------------------------------------------------------
__global__ __launch_bounds__(256)
void gnn_gemm_kernel(const float* __restrict__ X, float* __restrict__ out,
                     const float* __restrict__ Wself, const float* __restrict__ bself,
                     const float* __restrict__ Wmsg,  const float* __restrict__ bmsg,
                     const float* __restrict__ deg)
{
    __shared__ float sX[16 * LDS_STRIDE];
    __shared__ float sG[16 * LDS_STRIDE];

    const int row0 = blockIdx.x * 16;
    const int tid  = threadIdx.x;

    // Stage the 16x128 X tile and 16x128 agg tile (from `out`) into LDS.
    // 16 rows * 32 float4 = 512 float4 per tile; 256 threads -> 2 each.
    for (int i = tid; i < 16 * 32; i += 256) {
        int r  = i >> 5;
        int c4 = i & 31;
        float4 x = ((const float4*)(X   + (size_t)(row0 + r) * D))[c4];
        float4 g = ((const float4*)(out + (size_t)(row0 + r) * D))[c4];
        ((float4*)(sX + r * LDS_STRIDE))[c4] = x;
        ((float4*)(sG + r * LDS_STRIDE))[c4] = g;
    }
    __syncthreads();

    const int lane = tid & 31;
    const int wave = tid >> 5;
    const int col0 = wave * 16;
    const int mn   = lane & 15;             // M for A-frag, N for B-frag
    const int koff = (lane >> 4) << 1;      // lanes 16-31 hold K+2,K+3

    const float* wS = Wself + (size_t)(col0 + mn) * D;  // B[k][n] = W[n][k]
    const float* wM = Wmsg  + (size_t)(col0 + mn) * D;
    const float* aS = sX + mn * LDS_STRIDE;
    const float* aM = sG + mn * LDS_STRIDE;

    v8f c = {};
    #pragma unroll
    for (int k = 0; k < D; k += 4) {
        v2f a = { aS[k + koff], aS[k + koff + 1] };
        v2f b = { wS[k + koff], wS[k + koff + 1] };
        c = __builtin_amdgcn_wmma_f32_16x16x4_f32(false, a, false, b,
                                                  (short)0, c, false, false);
    }
    #pragma unroll
    for (int k = 0; k < D; k += 4) {
        v2f a = { aM[k + koff], aM[k + koff + 1] };
        v2f b = { wM[k + koff], wM[k + koff + 1] };
        c = __builtin_amdgcn_wmma_f32_16x16x4_f32(false, a, false, b,
                                                  (short)0, c, false, false);
    }

    // C/D layout: VGPR v -> M = v + 8*(lane>=16), N = lane&15
    const int col  = col0 + mn;
    const float bS = bself[col];
    const float bM = bmsg[col];
    const int rbase = row0 + ((lane >> 4) << 3);
    #pragma unroll
    for (int v = 0; v < 8; ++v) {
        int row = rbase + v;
        out[(size_t)row * D + col] = c[v] + bS + deg[row] * bM;
    }
}

// ---------------------------------------------------------------------------
extern "C" void kernel_launch(void* const* d_in, const int* in_sizes, int n_in,
                              void* d_out, int out_size, void* d_ws, size_t ws_size,
                              hipStream_t stream)
{
    const float* UF   = (const float*)d_in[0];
    const float* IF   = (const float*)d_in[1];
    const int*   edge = (const int*)d_in[2];
    const float* Wu   = (const float*)d_in[3];
    const float* bu   = (const float*)d_in[4];
    const float* Wi   = (const float*)d_in[5];
    const float* bi   = (const float*)d_in[6];
    const float* Wu2i = (const float*)d_in[7];
    const float* bu2i = (const float*)d_in[8];
    const float* Wi2u = (const float*)d_in[9];
    const float* bi2u = (const float*)d_in[10];

    const int U = in_sizes[0] / D;   // 100000
    const int I = in_sizes[1] / D;   // 50000
    const int E = in_sizes[2] / 2;   // 600000

    float* outU = (float*)d_out;                 // [U,D] — doubles as aggU
    float* outI = outU + (size_t)U * D;          // [I,D] — doubles as aggI
    float* degU = (float*)d_ws;                  // [U]
    float* degI = degU + U;                      // [I]

    // 1) zero aggregates (in d_out) and degrees (in d_ws)
    {
        long n4 = (long)(U + I) * D / 4;
        zero4_kernel<<<(unsigned)((n4 + 255) / 256), 256, 0, stream>>>((float4*)d_out, n4);
        long d4 = (long)(U + I) / 4;
        zero4_kernel<<<(unsigned)((d4 + 255) / 256), 256, 0, stream>>>((float4*)d_ws, d4);
    }

    // 2) scatter raw features + degree counts over edges
    {
        long threads = (long)E * 32;
        gnn_scatter_kernel<<<(unsigned)((threads + 255) / 256), 256, 0, stream>>>(
            UF, IF, edge, edge + E, outU, outI, degU, degI, E);
    }

    // 3) fused GEMMs (in-place over the aggregate buffers)
    gnn_gemm_kernel<<<U / 16, 256, 0, stream>>>(UF, outU, Wu, bu, Wi2u, bi2u, degU);
    gnn_gemm_kernel<<<I / 16, 256, 0, stream>>>(IF, outI, Wi, bi, Wu2i, bu2i, degI);
}